// GATConv_82978768158887
// MI455X (gfx1250) — compile-verified
//
#include <hip/hip_runtime.h>
#include <hip/hip_bf16.h>
#include <cstddef>

typedef __attribute__((ext_vector_type(2))) float v2f;
typedef __attribute__((ext_vector_type(8))) float v8f;

#define IN_CH   128
#define OUT_CH  64
#define NEG_SLOPE 0.2f
#define EPS_F   1e-10f

// ---------------------------------------------------------------------------
// h = x @ W.T via V_WMMA_F32_16X16X4_F32.  One wave per 16x16 output tile.
// A (16x4 f32): lane L<16 holds x[mBase+L][K+0..1], lane L>=16 holds K+2..3.
// B (4x16 f32): B[k][n] = W[n][k]; lane L holds W[nBase+(L&15)][K+2*(L>>4)+{0,1}].
// D (16x16 f32): VGPR r -> row mBase + r + 8*(L>>4), col nBase + (L&15).
// ---------------------------------------------------------------------------
__global__ void gat_gemm_wmma(const float* __restrict__ x,
                              const float* __restrict__ W,
                              float* __restrict__ h) {
    const int lane  = threadIdx.x & 31;
    const int wave  = threadIdx.x >> 5;
    const int tile  = blockIdx.x * (blockDim.x >> 5) + wave;
    const int mBase = (tile >> 2) * 16;     // 4 N-tiles per M-tile
    const int nBase = (tile & 3) * 16;

    const int l15  = lane & 15;
    const int kOff = (lane >> 4) << 1;      // 0 or 2

    const float* xRow = x + (size_t)(mBase + l15) * IN_CH + kOff;
    const float* wRow = W + (size_t)(nBase + l15) * IN_CH + kOff;

    v8f acc = {};
#pragma unroll
    for (int k = 0; k < IN_CH; k += 4) {
        v2f a = *(const v2f*)(xRow + k);
        v2f b = *(const v2f*)(wRow + k);
        acc = __builtin_amdgcn_wmma_f32_16x16x4_f32(
            /*neg_a=*/false, a, /*neg_b=*/false, b,
            /*c_mod=*/(short)0, acc, /*reuse_a=*/false, /*reuse_b=*/false);
    }

    const int rowHalf = (lane >> 4) << 3;   // 0 or 8
    float* o = h + (size_t)(mBase + rowHalf) * OUT_CH + nBase + l15;
#pragma unroll
    for (int r = 0; r < 8; ++r)
        o[(size_t)r * OUT_CH] = acc[r];
}

// Scalar fallback for a non-multiple-of-16 row tail (not hit for N=100000).
__global__ void gat_gemm_tail(const float* __restrict__ x,
                              const float* __restrict__ W,
                              float* __restrict__ h,
                              int rowStart, int nNodes) {
    int idx = blockIdx.x * blockDim.x + threadIdx.x;
    int row = rowStart + (idx >> 6);
    int col = idx & 63;
    if (row >= nNodes) return;
    const float* xr = x + (size_t)row * IN_CH;
    const float* wr = W + (size_t)col * IN_CH;
    float s = 0.f;
    for (int k = 0; k < IN_CH; ++k) s += xr[k] * wr[k];
    h[(size_t)row * OUT_CH + col] = s;
}

// ---------------------------------------------------------------------------
// e_l[n] = h[n,:]·a_l ; e_r[n] = h[n,:]·a_r.  One wave per node, float2/lane.
// ---------------------------------------------------------------------------
__global__ void gat_el_er(const float* __restrict__ h,
                          const float* __restrict__ a_l,
                          const float* __restrict__ a_r,
                          float* __restrict__ e_l,
                          float* __restrict__ e_r, int nNodes) {
    const int lane = threadIdx.x & 31;
    const int node = (blockIdx.x * blockDim.x + threadIdx.x) >> 5;
    if (node >= nNodes) return;
    float2 hv = ((const float2*)(h + (size_t)node * OUT_CH))[lane];
    float2 al = ((const float2*)a_l)[lane];
    float2 ar = ((const float2*)a_r)[lane];
    float pl = hv.x * al.x + hv.y * al.y;
    float pr = hv.x * ar.x + hv.y * ar.y;
#pragma unroll
    for (int off = 16; off > 0; off >>= 1) {
        pl += __shfl_xor(pl, off, 32);
        pr += __shfl_xor(pr, off, 32);
    }
    if (lane == 0) { e_l[node] = pl; e_r[node] = pr; }
}

// out[n,c] = bias[c]; node_max = -inf; node_sum = 0
__global__ void gat_init(float* __restrict__ out, const float* __restrict__ bias,
                         float* __restrict__ node_max, float* __restrict__ node_sum,
                         int nNodes) {
    int i = blockIdx.x * blockDim.x + threadIdx.x;
    if (i < nNodes * OUT_CH) out[i] = bias[i & (OUT_CH - 1)];
    if (i < nNodes) { node_max[i] = -__builtin_inff(); node_sum[i] = 0.f; }
}

// float atomic max via int-ordering trick (correct for mixed signs, -inf init)
__device__ __forceinline__ void atomicMaxF32(float* addr, float v) {
    if (v >= 0.f)
        atomicMax((int*)addr, __float_as_int(v));
    else
        atomicMin((unsigned int*)addr, __float_as_uint(v));
}

// pass 1: raw attn = leakyrelu(e_l[dst] + e_r[src]); segment max over dst
__global__ void gat_edge_max(const int* __restrict__ src, const int* __restrict__ dst,
                             const float* __restrict__ e_l, const float* __restrict__ e_r,
                             float* __restrict__ edge_w, float* __restrict__ node_max,
                             int nEdges) {
    int e = blockIdx.x * blockDim.x + threadIdx.x;
    if (e >= nEdges) return;
    int s = src[e], d = dst[e];
    float a = e_l[d] + e_r[s];
    a = (a > 0.f) ? a : NEG_SLOPE * a;
    edge_w[e] = a;
    atomicMaxF32(node_max + d, a);
}

// pass 2: w = exp(attn - max[dst]); segment sum over dst
__global__ void gat_edge_exp(const int* __restrict__ dst,
                             float* __restrict__ edge_w,
                             const float* __restrict__ node_max,
                             float* __restrict__ node_sum, int nEdges) {
    int e = blockIdx.x * blockDim.x + threadIdx.x;
    if (e >= nEdges) return;
    int d = dst[e];
    float w = __expf(edge_w[e] - node_max[d]);
    edge_w[e] = w;
    atomicAdd(node_sum + d, w);
}

// pass 3: out[dst,:] += h[src,:] * (w / (sum[dst]+eps)).  One wave per edge,
// float2 coalesced gather + 2 global f32 atomic adds per lane.
__global__ void gat_scatter(const int* __restrict__ src, const int* __restrict__ dst,
                            const float* __restrict__ h,
                            const float* __restrict__ edge_w,
                            const float* __restrict__ node_sum,
                            float* __restrict__ out, int nEdges) {
    const int lane = threadIdx.x & 31;
    const int e = (blockIdx.x * blockDim.x + threadIdx.x) >> 5;
    if (e >= nEdges) return;
    const int s = src[e], d = dst[e];
    const float coef = edge_w[e] / (node_sum[d] + EPS_F);
    float2 hv = ((const float2*)(h + (size_t)s * OUT_CH))[lane];
    float* o = out + (size_t)d * OUT_CH + lane * 2;
    atomicAdd(o + 0, hv.x * coef);
    atomicAdd(o + 1, hv.y * coef);
}

// ---------------------------------------------------------------------------
extern "C" void kernel_launch(void* const* d_in, const int* in_sizes, int n_in,
                              void* d_out, int out_size, void* d_ws, size_t ws_size,
                              hipStream_t stream) {
    const float* x    = (const float*)d_in[0];
    const int*   ei   = (const int*)d_in[1];
    const float* W    = (const float*)d_in[2];
    const float* a_l  = (const float*)d_in[3];
    const float* a_r  = (const float*)d_in[4];
    const float* bias = (const float*)d_in[5];
    float* out = (float*)d_out;

    const int nNodes = in_sizes[0] / IN_CH;   // 100000
    const int nEdges = in_sizes[1] / 2;       // 1600000
    const int* src = ei;
    const int* dst = ei + nEdges;

    // workspace layout (floats): h | e_l | e_r | node_max | node_sum | edge_w
    float* ws       = (float*)d_ws;
    float* h        = ws;
    float* e_l      = h + (size_t)nNodes * OUT_CH;
    float* e_r      = e_l + nNodes;
    float* node_max = e_r + nNodes;
    float* node_sum = node_max + nNodes;
    float* edge_w   = node_sum + nNodes;

    // --- GEMM (WMMA), 4 waves / block, one 16x16 tile / wave ---
    const int mFull = nNodes & ~15;
    const int tiles = (mFull / 16) * (OUT_CH / 16);
    if (tiles > 0)
        gat_gemm_wmma<<<(tiles + 3) / 4, 128, 0, stream>>>(x, W, h);
    if (mFull < nNodes) {
        int tailThreads = (nNodes - mFull) * OUT_CH;
        gat_gemm_tail<<<(tailThreads + 255) / 256, 256, 0, stream>>>(x, W, h, mFull, nNodes);
    }

    // --- attention scalars ---
    gat_el_er<<<(nNodes * 32 + 255) / 256, 256, 0, stream>>>(h, a_l, a_r, e_l, e_r, nNodes);

    // --- init out(+bias), max, sum ---
    int initN = nNodes * OUT_CH;
    gat_init<<<(initN + 255) / 256, 256, 0, stream>>>(out, bias, node_max, node_sum, nNodes);

    // --- segment softmax + scatter ---
    gat_edge_max<<<(nEdges + 255) / 256, 256, 0, stream>>>(src, dst, e_l, e_r, edge_w, node_max, nEdges);
    gat_edge_exp<<<(nEdges + 255) / 256, 256, 0, stream>>>(dst, edge_w, node_max, node_sum, nEdges);
    gat_scatter<<<((size_t)nEdges * 32 + 255) / 256, 256, 0, stream>>>(src, dst, h, edge_w, node_sum, out, nEdges);
}